// TTFLoss_20203526160999
// MI455X (gfx1250) — compile-verified
//
#include <hip/hip_runtime.h>
#include <math.h>

// Problem constants (match reference)
#define B_ 16
#define K_ 128
#define C_ 80
#define H_ 128
#define W_ 128
#define ALPHA_ 0.54f

// Box descriptor: 16 floats per (image, sorted-box)
//  [0] active(int) [1] cls(int) [2] cx(int) [3] cy(int) [4] hr(int) [5] wr(int)
//  [6] 1/(2*sy^2)  [7] 1/(2*sx^2) [8] factor/div  [9..12] x1,y1,x2,y2 (input coords)
#define BOX_F 16
#define N_BOX_F (B_ * K_ * BOX_F)
#define NBLK2 (B_ * 16)              // wh-loss: 16 blocks/image, 256 thr, 4 px/thr
#define NBLK3 (B_ * C_ * 16)         // focal: one (b,c) plane = 16 blocks, 4 px/thr
#define OFF_P2 (N_BOX_F)
#define OFF_P3 (OFF_P2 + NBLK2 * 2)
// total ws floats = OFF_P3 + NBLK3*3 = 94720 floats (~370 KB)

#if defined(__AMDGCN__) && __has_builtin(__builtin_amdgcn_global_load_async_to_lds_b128) && __has_builtin(__builtin_amdgcn_s_wait_asynccnt)
#define USE_ASYNC_LDS 1
typedef __attribute__((__vector_size__(4 * sizeof(int)))) int v4i_t;
typedef __attribute__((address_space(1))) v4i_t gv4_t;   // global v4i
typedef __attribute__((address_space(3))) v4i_t lv4_t;   // LDS v4i
#endif

__device__ __forceinline__ float blockReduce256(float v, float* red) {
  int t = threadIdx.x;
  __syncthreads();                 // protect red against previous use
  red[t] = v;
  __syncthreads();
  for (int s = 128; s > 0; s >>= 1) {
    if (t < s) red[t] += red[t + s];
    __syncthreads();
  }
  return red[0];
}

// ---------------- Kernel 1: per-image box prep (sort + gaussian params) -------
__global__ __launch_bounds__(K_) void k_prep(const float* __restrict__ target,
                                             const int* __restrict__ mask,
                                             float* __restrict__ boxws) {
  int b = blockIdx.x;
  int k = threadIdx.x;
  const float* t = target + (b * K_ + k) * 5;
  float clsf = t[0];
  float x1 = t[1], y1 = t[2], x2 = t[3], y2 = t[4];
  bool valid = mask[b * K_ + k] != 0;
  float key = valid ? logf((x2 - x1) * (y2 - y1)) : -INFINITY;

  __shared__ float keys[K_];
  keys[k] = key;
  __syncthreads();
  // stable descending rank == argsort(-key) position
  int rank = 0;
  float al0 = -INFINITY;
  for (int j = 0; j < K_; ++j) {
    float kj = keys[j];
    if (kj > key || (kj == key && j < k)) rank++;
    al0 = fmaxf(al0, kj);
  }
  if (al0 == -INFINITY) al0 = 0.f;   // all invalid -> al[0] = 0
  float al = valid ? key : 0.f;
  float factor = 2.f - al / (al0 + 1e-7f);

  float fx1 = fminf(fmaxf(x1 * 0.25f, 0.f), (float)(W_ - 1));
  float fy1 = fminf(fmaxf(y1 * 0.25f, 0.f), (float)(H_ - 1));
  float fx2 = fminf(fmaxf(x2 * 0.25f, 0.f), (float)(W_ - 1));
  float fy2 = fminf(fmaxf(y2 * 0.25f, 0.f), (float)(H_ - 1));
  float feat_h = fy2 - fy1, feat_w = fx2 - fx1;
  int cx = (int)((x1 + x2) * 0.5f * 0.25f);
  int cy = (int)((y1 + y2) * 0.5f * 0.25f);
  int hr = (int)(feat_h * 0.5f * ALPHA_);
  int wr = (int)(feat_w * 0.5f * ALPHA_);
  float sy = (float)(2 * hr + 1) / 6.f;
  float sx = (float)(2 * wr + 1) / 6.f;
  float i2sy = 1.f / (2.f * sy * sy);
  float i2sx = 1.f / (2.f * sx * sx);
  bool inb = (cx >= 0) && (cx < W_) && (cy >= 0) && (cy < H_);
  int active = (valid && inb) ? 1 : 0;

  // div = G.sum = (sum_y gy) * (sum_x gx), separable
  float Sy = 0.f, Sx = 0.f;
  int y0 = cy - hr; if (y0 < 0) y0 = 0;
  int y1i = cy + hr; if (y1i > H_ - 1) y1i = H_ - 1;
  for (int y = y0; y <= y1i; ++y) { float dy = (float)(y - cy); Sy += __expf(-dy * dy * i2sy); }
  int x0 = cx - wr; if (x0 < 0) x0 = 0;
  int x1i = cx + wr; if (x1i > W_ - 1) x1i = W_ - 1;
  for (int x = x0; x <= x1i; ++x) { float dx = (float)(x - cx); Sx += __expf(-dx * dx * i2sx); }
  float coef = factor / (Sy * Sx + 1e-7f);

  float* o = boxws + (b * K_ + rank) * BOX_F;   // scatter into sorted slot
  o[0] = __int_as_float(active);
  o[1] = __int_as_float((int)clsf);
  o[2] = __int_as_float(cx);
  o[3] = __int_as_float(cy);
  o[4] = __int_as_float(hr);
  o[5] = __int_as_float(wr);
  o[6] = i2sy; o[7] = i2sx; o[8] = coef;
  o[9] = x1; o[10] = y1; o[11] = x2; o[12] = y2;
  o[13] = 0.f; o[14] = 0.f; o[15] = 0.f;
}

// ---------------- Kernel 2: wh IoU loss (winner box, 4 px/thread) -------------
__global__ __launch_bounds__(256) void k_whloss(const float* __restrict__ wh,
                                                const float* __restrict__ boxws,
                                                float* __restrict__ p2) {
  __shared__ float boxl[K_ * BOX_F];   // 8 KB box table
  __shared__ float red[256];
  int blk = blockIdx.x;
  int b = blk >> 4;
  int po4 = ((blk & 15) << 8) + threadIdx.x;  // group of 4 pixels
  int px = po4 << 2;
  int y = px >> 7;
  int x0 = px & (W_ - 1);                     // multiple of 4 -> 16B aligned
  const float* g = boxws + b * K_ * BOX_F;

#ifdef USE_ASYNC_LDS
  // gfx1250 async global->LDS staging (ASYNCcnt tracked)
  for (int i = threadIdx.x; i < (K_ * BOX_F) / 4; i += 256) {
    float* gp = const_cast<float*>(g + i * 4);
    __builtin_amdgcn_global_load_async_to_lds_b128((gv4_t*)gp, (lv4_t*)&boxl[i * 4], 0, 0);
  }
  __builtin_amdgcn_s_wait_asynccnt(0);
#else
  for (int i = threadIdx.x; i < K_ * BOX_F; i += 256) boxl[i] = g[i];
#endif
  __syncthreads();

  int w[4] = {-1, -1, -1, -1};
  float gw[4] = {0.f, 0.f, 0.f, 0.f};
  for (int s = 0; s < K_; ++s) {
    const float* bx = &boxl[s * BOX_F];
    if (__float_as_int(bx[0]) == 0) continue;
    int dy = y - __float_as_int(bx[3]);
    int ady = dy < 0 ? -dy : dy;
    if (ady > __float_as_int(bx[4])) continue;
    int cxv = __float_as_int(bx[2]);
    int wrv = __float_as_int(bx[5]);
    float ey = __expf(-(float)(dy * dy) * bx[6]);
    float i2sx = bx[7];
#pragma unroll
    for (int j = 0; j < 4; ++j) {
      int dx = x0 + j - cxv;
      int adx = dx < 0 ? -dx : dx;
      if (adx <= wrv) {
        w[j] = s;   // last (smallest area) wins == max sorted index
        gw[j] = ey * __expf(-(float)(dx * dx) * i2sx);
      }
    }
  }

  // wh preds: 4 comps x 4 pixels via aligned float4 loads
  int base = (b * 4 * H_ + y) * W_ + x0;
  float4 pl4 = *(const float4*)(wh + base);
  float4 pt4 = *(const float4*)(wh + base + H_ * W_);
  float4 pr4 = *(const float4*)(wh + base + 2 * H_ * W_);
  float4 pb4 = *(const float4*)(wh + base + 3 * H_ * W_);
  float plv[4] = {pl4.x, pl4.y, pl4.z, pl4.w};
  float ptv[4] = {pt4.x, pt4.y, pt4.z, pt4.w};
  float prv[4] = {pr4.x, pr4.y, pr4.z, pr4.w};
  float pbv[4] = {pb4.x, pb4.y, pb4.z, pb4.w};

  float lw = 0.f, rw = 0.f;
  float yf = (float)(y * 4);
#pragma unroll
  for (int j = 0; j < 4; ++j) {
    if (w[j] >= 0) {
      const float* bx = &boxl[w[j] * BOX_F];
      float rwj = gw[j] * bx[8];
      float xf = (float)((x0 + j) * 4);
      float tl = xf - bx[9];
      float tt = yf - bx[10];
      float tr = bx[11] - xf;
      float tb = bx[12] - yf;
      float pl = plv[j] * 16.f;
      float pt = ptv[j] * 16.f;
      float pr = prv[j] * 16.f;
      float pb = pbv[j] * 16.f;
      float ta = (tl + tr) * (tt + tb);
      float pa = (pl + pr) * (pt + pb);
      float wi = fminf(pl, tl) + fminf(pr, tr);
      float hi = fminf(pt, tt) + fminf(pb, tb);
      float ai = wi * hi;
      float au = ta + pa - ai;
      float iou = (ai + 1.f) / (au + 1.f);
      lw += (1.f - iou) * rwj;
      rw += rwj;
    }
  }
  float s1 = blockReduce256(lw, red);
  float s2 = blockReduce256(rw, red);
  if (threadIdx.x == 0) { p2[blk * 2 + 0] = s1; p2[blk * 2 + 1] = s2; }
}

// ------------- Kernel 3: focal loss over hm (84 MB stream, 4 px/thread) ------
__global__ __launch_bounds__(256) void k_focal(const float* __restrict__ hm,
                                               const float* __restrict__ boxws,
                                               float* __restrict__ p3) {
  __shared__ float red[256];
  int blk = blockIdx.x;
  int plane = blk >> 4;          // b*C + c
  int b = plane / C_;
  int c = plane - b * C_;
  int po4 = ((blk & 15) << 8) + threadIdx.x;
  int px = po4 << 2;
  int y = px >> 7;
  int x0 = px & (W_ - 1);

  float4 hv = *(const float4*)(hm + (size_t)plane * (H_ * W_) + px);
  float pv[4] = {hv.x, hv.y, hv.z, hv.w};

  // heatmap value = max gaussian among active boxes of class c covering pixel
  const float* g = boxws + b * K_ * BOX_F;
  float h[4] = {0.f, 0.f, 0.f, 0.f};
  for (int s = 0; s < K_; ++s) {
    const float* bx = g + s * BOX_F;
    if (__float_as_int(bx[0]) == 0) continue;
    if (__float_as_int(bx[1]) != c) continue;   // block-uniform reject (1/80 hit)
    int dy = y - __float_as_int(bx[3]);
    int ady = dy < 0 ? -dy : dy;
    if (ady > __float_as_int(bx[4])) continue;
    int cxv = __float_as_int(bx[2]);
    int wrv = __float_as_int(bx[5]);
    float ey = __expf(-(float)(dy * dy) * bx[6]);
    float i2sx = bx[7];
#pragma unroll
    for (int j = 0; j < 4; ++j) {
      int dx = x0 + j - cxv;
      int adx = dx < 0 ? -dx : dx;
      if (adx <= wrv)
        h[j] = fmaxf(h[j], ey * __expf(-(float)(dx * dx) * i2sx));
    }
  }

  float posl = 0.f, negl = 0.f, np = 0.f;
#pragma unroll
  for (int j = 0; j < 4; ++j) {
    float pred = 1.f / (1.f + __expf(-pv[j]));
    pred = fminf(fmaxf(pred, 1e-4f), 1.f - 1e-4f);
    if (h[j] == 1.0f) {          // exact: exp(0)*exp(0)==1 at box centers
      float om = 1.f - pred;
      posl += __logf(pred) * om * om;
      np += 1.f;
    } else {
      float oh = 1.f - h[j];
      float oh2 = oh * oh;
      negl += __logf(1.f - pred) * pred * pred * oh2 * oh2;
    }
  }
  float s1 = blockReduce256(posl, red);
  float s2 = blockReduce256(negl, red);
  float s3 = blockReduce256(np, red);
  if (threadIdx.x == 0) {
    p3[blk * 3 + 0] = s1; p3[blk * 3 + 1] = s2; p3[blk * 3 + 2] = s3;
  }
}

// ---------------- Kernel 4: deterministic final reduce + combine -------------
__global__ __launch_bounds__(256) void k_final(const float* __restrict__ p2,
                                               const float* __restrict__ p3,
                                               float* __restrict__ out) {
  __shared__ float red[256];
  int t = threadIdx.x;
  float lw = 0.f, rw = 0.f;
  for (int i = t; i < NBLK2; i += 256) { lw += p2[i * 2 + 0]; rw += p2[i * 2 + 1]; }
  float pos = 0.f, neg = 0.f, np = 0.f;
  for (int i = t; i < NBLK3; i += 256) {
    pos += p3[i * 3 + 0]; neg += p3[i * 3 + 1]; np += p3[i * 3 + 2];
  }
  lw = blockReduce256(lw, red);
  rw = blockReduce256(rw, red);
  pos = blockReduce256(pos, red);
  neg = blockReduce256(neg, red);
  np = blockReduce256(np, red);
  if (t == 0) {
    float hm_loss = (np > 0.f) ? (-(pos + neg) / fmaxf(np, 1.f)) : (-neg);
    float wh_loss = lw / fmaxf(rw, 1.f);
    out[0] = hm_loss + wh_loss;
  }
}

extern "C" void kernel_launch(void* const* d_in, const int* in_sizes, int n_in,
                              void* d_out, int out_size, void* d_ws, size_t ws_size,
                              hipStream_t stream) {
  const float* hm = (const float*)d_in[0];
  const float* wh = (const float*)d_in[1];
  const float* target = (const float*)d_in[2];
  const int* mask = (const int*)d_in[3];
  float* ws = (float*)d_ws;
  float* boxws = ws;
  float* p2 = ws + OFF_P2;
  float* p3 = ws + OFF_P3;

  k_prep<<<B_, K_, 0, stream>>>(target, mask, boxws);
  k_whloss<<<NBLK2, 256, 0, stream>>>(wh, boxws, p2);
  k_focal<<<NBLK3, 256, 0, stream>>>(hm, boxws, p3);
  k_final<<<1, 256, 0, stream>>>(p2, p3, (float*)d_out);
}